// E79CoupledMatrixCell_9165460210268
// MI455X (gfx1250) — compile-verified
//
#include <hip/hip_runtime.h>
#include <hip/hip_bf16.h>
#include <math.h>

typedef __attribute__((ext_vector_type(2))) float v2f;
typedef __attribute__((ext_vector_type(8))) float v8f;

#define DIMK   1024
#define NOUT   256          // 4 * N_STATE
#define NS     64           // N_STATE
#define TT     2048
#define BB     16
#define TBROWS (TT * BB)    // 32768 GEMM rows
#define EPSV   1e-6f
#define LDA    1028         // padded LDS stride (1024+4): 64 dwords/fetch hit 64 distinct banks
#define LDSM   65           // padded state stride: row & column access both conflict-free
#define SL     16           // row-slice length per thread in scan (NS/4)

__device__ __forceinline__ float sigmoidf_(float x) {
    return 1.0f / (1.0f + expf(-x));
}

// Reduce across the 4 slice-owners of one row (adjacent lanes: xor 1, xor 2).
__device__ __forceinline__ float red4_(float v) {
    v += __shfl_xor(v, 1, 32);
    v += __shfl_xor(v, 2, 32);
    return v;
}

// ---------------------------------------------------------------------------
// Phase 1: proj[t*B+b][n] = sum_d x[t,b,d] * W[n,d]   via V_WMMA_F32_16X16X4_F32
// Block = 512 threads (16 waves). Block computes 16 rows x 256 cols.
// ---------------------------------------------------------------------------
__global__ __launch_bounds__(512)
void proj_gemm_kernel(const float* __restrict__ x,
                      const float* __restrict__ W,
                      float* __restrict__ proj) {
    __shared__ float Atile[16 * LDA];

    const int tid  = threadIdx.x;
    const int row0 = blockIdx.x * 16;

    for (int it = tid; it < 16 * (DIMK / 4); it += 512) {
        const int r  = it / (DIMK / 4);
        const int c4 = it % (DIMK / 4);
        const float4 v =
            ((const float4*)(x + (size_t)(row0 + r) * DIMK))[c4];
        float* dst = &Atile[r * LDA + c4 * 4];
        dst[0] = v.x; dst[1] = v.y; dst[2] = v.z; dst[3] = v.w;
    }
    __syncthreads();

    const int wave = tid >> 5;       // 0..15 -> N tile
    const int lane = tid & 31;
    const int m    = lane & 15;
    const int h    = lane >> 4;      // K half-select
    const int ncol = wave * 16 + m;  // output column == W row

    const float* wrow = W + (size_t)ncol * DIMK;

    v8f acc = {};
    #pragma unroll 8
    for (int kk = 0; kk < DIMK; kk += 4) {
        const float* ap = &Atile[m * LDA + kk + 2 * h];
        v2f a; a.x = ap[0]; a.y = ap[1];
        v2f b; b.x = wrow[kk + 2 * h]; b.y = wrow[kk + 2 * h + 1];
        acc = __builtin_amdgcn_wmma_f32_16x16x4_f32(
            false, a, false, b, (short)0, acc, false, false);
    }

    #pragma unroll
    for (int j = 0; j < 8; ++j) {
        const int r = row0 + j + 8 * h;
        proj[(size_t)r * NOUT + ncol] = acc[j];
    }
}

// ---------------------------------------------------------------------------
// Phase 2: sequential gated scan. One block per batch, 256 threads (8 waves).
// Thread (i = tid>>2, g = tid&3) owns S[i][16g..16g+16) and M[i][16g..16g+16)
// in VGPRs; padded LDS mirrors serve the transposed (column) matvecs.
// ---------------------------------------------------------------------------
__global__ __launch_bounds__(256)
void scan_kernel(const float* __restrict__ proj,
                 const float* __restrict__ S0,
                 const float* __restrict__ M0,
                 const float* __restrict__ b_s,
                 const float* __restrict__ b_m,
                 float* __restrict__ out,
                 float* __restrict__ Sf,
                 float* __restrict__ Mf) {
    __shared__ float Sl[NS * LDSM];
    __shared__ float Ml[NS * LDSM];
    __shared__ float buf[NOUT];          // [k(64) | v(64) | q(64) | m(64)]
    __shared__ float scol[NS], mcol[NS];
    __shared__ float red[8];

    float* kbuf = buf;
    float* vbuf = buf + NS;
    float* qbuf = buf + 2 * NS;
    float* mbuf = buf + 3 * NS;

    const int b   = blockIdx.x;
    const int tid = threadIdx.x;
    const int i   = tid >> 2;            // row 0..63
    const int g   = tid & 3;             // slice 0..3
    const int c0  = g * SL;              // first column of slice
    const int w   = tid >> 5;            // wave id 0..7
    const int lane = tid & 31;

    float Srow[SL], Mrow[SL];
    #pragma unroll
    for (int l = 0; l < SL; ++l) {
        const float s  = S0[((size_t)b * NS + i) * NS + c0 + l];
        const float mm = M0[((size_t)b * NS + i) * NS + c0 + l];
        Srow[l] = s;  Mrow[l] = mm;
        Sl[i * LDSM + c0 + l] = s;
        Ml[i * LDSM + c0 + l] = mm;
    }
    const float bsi = b_s[i];
    const float bmi = b_m[i];

    // Software pipeline: each thread owns exactly one float of the 256-float
    // proj row; load for t is issued one iteration ahead.
    const float* pbase = proj + (size_t)b * NOUT;
    float pv = pbase[tid];               // t = 0
    __syncthreads();

    for (int t = 0; t < TT; ++t) {
        // Issue next step's load + deep prefetch immediately (hidden by step t).
        float pv_next = 0.0f;
        if (t + 1 < TT) pv_next = pbase[(size_t)(t + 1) * BB * NOUT + tid];
        if (t + 4 < TT)
            __builtin_prefetch(&pbase[(size_t)(t + 4) * BB * NOUT + tid], 0, 1);

        buf[tid] = pv;

        // Norm partials: waves 0,1 hold k; waves 6,7 hold m.
        float sq = pv * pv;
        #pragma unroll
        for (int off = 16; off >= 1; off >>= 1) sq += __shfl_xor(sq, off, 32);
        if (lane == 0) red[w] = sq;
        __syncthreads();

        const float inv_k = 1.0f / (sqrtf(red[0] + red[1]) + EPSV);
        const float inv_m = 1.0f / (sqrtf(red[6] + red[7]) + EPSV);

        // --- M k (row), M^T k (col), S k (row): partials over this slice ---
        float amk = 0.0f, amkT = 0.0f, ask = 0.0f;
        #pragma unroll
        for (int l = 0; l < SL; ++l) {
            const int j = c0 + l;
            const float kj = kbuf[j];                  // broadcast
            amk  += Mrow[l] * kj;
            ask  += Srow[l] * kj;
            amkT += Ml[j * LDSM + i] * kj;             // column, conflict-free
        }
        amk  = red4_(amk);
        ask  = red4_(ask);
        amkT = red4_(amkT);

        const float s_row   = sigmoidf_(amk * inv_k + bsi);
        const float s_colv  = sigmoidf_(amkT * inv_k + bsi);
        const float s_delta = vbuf[i] - ask * inv_k;
        if (g == 0) scol[i] = s_colv;
        __syncthreads();

        // --- S update, fused with S_new·q and S_new·m_norm row partials ---
        float asq = 0.0f, asm_ = 0.0f;
        #pragma unroll
        for (int l = 0; l < SL; ++l) {
            const int j = c0 + l;
            const float snew = s_row * Srow[l] * scol[j]
                             + s_delta * (kbuf[j] * inv_k);
            Srow[l] = snew;
            Sl[i * LDSM + j] = snew;
            asq  += snew * qbuf[j];
            asm_ += snew * (mbuf[j] * inv_m);
        }
        asq  = red4_(asq);
        asm_ = red4_(asm_);
        __syncthreads();

        // --- S_new^T m_norm (col) and M m_norm (row) partials ---
        float asmT = 0.0f, amret = 0.0f;
        #pragma unroll
        for (int l = 0; l < SL; ++l) {
            const int j = c0 + l;
            const float mj = mbuf[j];
            asmT  += Sl[j * LDSM + i] * mj;            // column of new S
            amret += Mrow[l] * mj;
        }
        asmT  = red4_(asmT);
        amret = red4_(amret);

        const float m_row   = sigmoidf_(asm_ + bmi);
        const float m_colv  = sigmoidf_(asmT * inv_m + bmi);
        const float m_delta = s_delta - amret * inv_m;
        if (g == 0) mcol[i] = m_colv;
        __syncthreads();

        // --- M update ---
        #pragma unroll
        for (int l = 0; l < SL; ++l) {
            const int j = c0 + l;
            const float mnew = m_row * Mrow[l] * mcol[j]
                             + m_delta * (mbuf[j] * inv_m);
            Mrow[l] = mnew;
            Ml[i * LDSM + j] = mnew;
        }

        // out = Sq^2 * sigmoid(Sq)
        if (g == 0) {
            const float Sq = asq;
            out[((size_t)t * BB + b) * NS + i] = Sq * Sq * sigmoidf_(Sq);
        }

        pv = pv_next;
        __syncthreads();   // Ml rows done before next step's column reads;
                           // buf/scol/mcol/red safe to overwrite
    }

    #pragma unroll
    for (int l = 0; l < SL; ++l) {
        Sf[((size_t)b * NS + i) * NS + c0 + l] = Srow[l];
        Mf[((size_t)b * NS + i) * NS + c0 + l] = Mrow[l];
    }
}

extern "C" void kernel_launch(void* const* d_in, const int* in_sizes, int n_in,
                              void* d_out, int out_size, void* d_ws, size_t ws_size,
                              hipStream_t stream) {
    const float* x   = (const float*)d_in[0];   // (T,B,1024)
    const float* S0  = (const float*)d_in[1];   // (B,64,64)
    const float* M0  = (const float*)d_in[2];   // (B,64,64)
    const float* W   = (const float*)d_in[3];   // (256,1024)
    const float* b_s = (const float*)d_in[4];   // (64,)
    const float* b_m = (const float*)d_in[5];   // (64,)

    float* out = (float*)d_out;                    // (T,B,64)
    float* Sf  = out + (size_t)TT * BB * NS;       // (B,64,64)
    float* Mf  = Sf + (size_t)BB * NS * NS;        // (B,64,64)

    float* proj = (float*)d_ws;                    // needs 32 MB scratch
    (void)in_sizes; (void)n_in; (void)out_size; (void)ws_size;

    proj_gemm_kernel<<<TBROWS / 16, 512, 0, stream>>>(x, W, proj);
    scan_kernel<<<BB, 256, 0, stream>>>(proj, S0, M0, b_s, b_m, out, Sf, Mf);
}